// GINet_4432406250029
// MI455X (gfx1250) — compile-verified
//
#include <hip/hip_runtime.h>
#include <hip/hip_bf16.h>

// ---------------------------------------------------------------------------
// GINE-style GNN forward for MI455X (gfx1250, wave32).
//
// Roofline: edge gather/scatter dominates (~300MB L2-resident traffic per
// layer); dense math is ~18 GFLOP total (negligible). Design:
//  - fused edge kernel: persistent waves, We cached in 48 VGPRs (launch
//    bounds chosen so it stays in registers, no spills), WMMA fp32
//    edge-linear, gather h[src], relu, atomic scatter-add to aggr[dst]
//  - node MLP kernel: persistent blocks, W0+W1 resident in LDS (staged via
//    gfx1250 GLOBAL_LOAD_ASYNC_TO_LDS_B128 / ASYNCcnt), two back-to-back
//    fp32 WMMA GEMMs per 16-node tile, graph pooling via atomics
//  - fp32 WMMA (V_WMMA_F32_16X16X4_F32) keeps reference precision.
// ---------------------------------------------------------------------------

typedef float v2f __attribute__((ext_vector_type(2)));
typedef float v8f __attribute__((ext_vector_type(8)));

#define DIM   96
#define EDIM  16
#define CATD  288
#define NGR   512

static __device__ __forceinline__ v8f wmma_f32(v2f a, v2f b, v8f c) {
  // D(16x16,f32) = A(16x4,f32) * B(4x16,f32) + C
  return __builtin_amdgcn_wmma_f32_16x16x4_f32(false, a, false, b, (short)0, c,
                                               false, false);
}

// gfx1250 async global->LDS copy, 128b per lane (tracked by ASYNCcnt)
static __device__ __forceinline__ void async_ld_lds_b128(void* lds,
                                                         const void* gptr) {
  unsigned ldsoff = (unsigned)(unsigned long long)lds;  // low 32b = LDS offset
  asm volatile("global_load_async_to_lds_b128 %0, %1, off"
               :
               : "v"(ldsoff), "v"(gptr)
               : "memory");
}
static __device__ __forceinline__ void wait_async0() {
  asm volatile("s_wait_asynccnt 0" ::: "memory");
}

// h_vn = h_in + vn_emb (broadcast over rows); aggr = 0
__global__ void prep_kernel(const float* __restrict__ hin,
                            const float* __restrict__ vn,
                            float* __restrict__ hvn,
                            float* __restrict__ aggr, int total) {
  int i = blockIdx.x * blockDim.x + threadIdx.x;
  if (i < total) {
    hvn[i]  = hin[i] + vn[i % DIM];
    aggr[i] = 0.0f;
  }
}

__global__ void zero_kernel(float* __restrict__ p, int total) {
  int i = blockIdx.x * blockDim.x + threadIdx.x;
  if (i < total) p[i] = 0.0f;
}

// ---------------------------------------------------------------------------
// Edge kernel: persistent waves; each iteration handles a tile of 16 edges.
//   e    = edge_attr(16x16) @ We(16x96)      -> 6 N-tiles x 4 K-steps, WMMA
//   msg  = relu(hvn[src] + e + be)
//   aggr[dst] += msg (atomic)
// We/be cached in registers across all tiles. Epilogue is r-outer so each
// edge row loads src/dst once; full tiles take the unguarded fast path.
// __launch_bounds__(256,1): ~130 live VGPRs/lane; occupancy 1 block/EU is
// fine (L2 atomic/gather bound) and avoids scratch spills of the B cache.
// A-layout (16x4 f32): lane m=lane&15 holds row m; V0={K+0|K+2}, V1={K+1|K+3}
// B-layout (4x16 f32): lane n=lane&15 holds col n; same K split per lane half
// C/D-layout: VGPR r -> rows {r | r+8}, lanes hold N.
// ---------------------------------------------------------------------------
__global__ __launch_bounds__(256, 1) void edge_kernel(
    const float* __restrict__ ea, const int* __restrict__ src,
    const int* __restrict__ dst, const float* __restrict__ We,
    const float* __restrict__ be, const float* __restrict__ hvn,
    float* __restrict__ aggr, int E, int numTiles) {
  const int lane = threadIdx.x & 31;
  const int m    = lane & 15;
  const int koff = (lane >> 4) << 1;  // 0 or 2
  const int half = (lane >> 4) << 3;  // C/D row offset 0 or 8

  // --- cache B (We) in registers: 4 K-steps x 6 N-tiles x v2f ---
  v2f Bw[4][6];
  float bias[6];
#pragma unroll
  for (int kk = 0; kk < 4; ++kk) {
    int k = 4 * kk + koff;
#pragma unroll
    for (int nt = 0; nt < 6; ++nt) {
      Bw[kk][nt].x = We[k * DIM + nt * 16 + m];
      Bw[kk][nt].y = We[(k + 1) * DIM + nt * 16 + m];
    }
  }
#pragma unroll
  for (int nt = 0; nt < 6; ++nt) bias[nt] = be[nt * 16 + m];

  const int wavesPerGrid = gridDim.x * (blockDim.x >> 5);
  for (int tile = blockIdx.x * (blockDim.x >> 5) + (threadIdx.x >> 5);
       tile < numTiles; tile += wavesPerGrid) {
    const int e0 = tile * 16;

    // --- load A: edge_attr tile (clamped; tail stores masked below) ---
    int erow = e0 + m;
    if (erow >= E) erow = E - 1;
    const float2* earow = (const float2*)(ea + erow * EDIM);
    v2f a[4];
#pragma unroll
    for (int kk = 0; kk < 4; ++kk) {
      float2 t = earow[(4 * kk + koff) >> 1];
      a[kk].x = t.x;
      a[kk].y = t.y;
    }

    // --- WMMA ---
    v8f acc[6];
#pragma unroll
    for (int nt = 0; nt < 6; ++nt) acc[nt] = {};
#pragma unroll
    for (int kk = 0; kk < 4; ++kk)
#pragma unroll
      for (int nt = 0; nt < 6; ++nt)
        acc[nt] = wmma_f32(a[kk], Bw[kk][nt], acc[nt]);

    // --- epilogue: + be + hvn[src], relu, atomic scatter-add ---
    if (e0 + 16 <= E) {  // wave-uniform fast path: no per-element guards
#pragma unroll
      for (int r = 0; r < 8; ++r) {
        const int e = e0 + r + half;
        const int s = src[e], d = dst[e];
        const float* hrow = hvn + (size_t)s * DIM + m;
        float*       arow = aggr + (size_t)d * DIM + m;
#pragma unroll
        for (int nt = 0; nt < 6; ++nt) {
          float v = acc[nt][r] + bias[nt] + hrow[nt * 16];
          v = v > 0.0f ? v : 0.0f;
          atomicAdd(&arow[nt * 16], v);
        }
      }
    } else {  // tail tile
#pragma unroll
      for (int r = 0; r < 8; ++r) {
        const int e = e0 + r + half;
        if (e < E) {
          const int s = src[e], d = dst[e];
          const float* hrow = hvn + (size_t)s * DIM + m;
          float*       arow = aggr + (size_t)d * DIM + m;
#pragma unroll
          for (int nt = 0; nt < 6; ++nt) {
            float v = acc[nt][r] + bias[nt] + hrow[nt * 16];
            v = v > 0.0f ? v : 0.0f;
            atomicAdd(&arow[nt * 16], v);
          }
        }
      }
    }
  }
}

// ---------------------------------------------------------------------------
// Node kernel: persistent blocks; W0 and W1 both resident in LDS (async
// staged once). Per 16-node tile (one wave):
//   z = relu((hvn + aggr) @ W0 + b0)   [WMMA]
//   o = z @ W1 + b1                    [WMMA, z staged in per-wave LDS to
//                                       convert D-layout -> A-layout]
//   h_out = o;  g[batch[n], layer*96+col] += o
// No inner barriers: z staging is wave-private (DS-cnt ordering suffices).
// ---------------------------------------------------------------------------
#define NODE_WAVES 4
__global__ __launch_bounds__(32 * NODE_WAVES, 1) void node_kernel(
    const float* __restrict__ hvn, const float* __restrict__ aggr,
    const float* __restrict__ W0, const float* __restrict__ b0,
    const float* __restrict__ W1, const float* __restrict__ b1,
    float* __restrict__ hout, float* __restrict__ g,
    const int* __restrict__ batch, int layer, int N, int numChunks) {
  __shared__ float W0lds[DIM * DIM];           // 36 KB
  __shared__ float W1lds[DIM * DIM];           // 36 KB
  __shared__ float zs[NODE_WAVES][16 * DIM];   // 24 KB  (96 KB total / WGP-LDS)

  const int lane = threadIdx.x & 31;
  const int wid  = threadIdx.x >> 5;
  const int m    = lane & 15;
  const int koff = (lane >> 4) << 1;
  const int half = (lane >> 4) << 3;

  // --- stage both weight matrices via async global->LDS (ASYNCcnt path) ---
  for (int i = threadIdx.x; i < (DIM * DIM) / 4; i += blockDim.x) {
    async_ld_lds_b128(&W0lds[i * 4], W0 + i * 4);
    async_ld_lds_b128(&W1lds[i * 4], W1 + i * 4);
  }
  wait_async0();
  __syncthreads();

  // biases cached in registers across chunks
  float b0c[6], b1c[6];
#pragma unroll
  for (int nt = 0; nt < 6; ++nt) {
    b0c[nt] = b0[nt * 16 + m];
    b1c[nt] = b1[nt * 16 + m];
  }

  for (int chunk = blockIdx.x; chunk < numChunks; chunk += gridDim.x) {
    const int n0 = (chunk * NODE_WAVES + wid) * 16;
    int arow = n0 + m;
    if (arow >= N) arow = N - 1;
    if (arow < 0) arow = 0;
    const float* ha = hvn + (size_t)arow * DIM;
    const float* ga = aggr + (size_t)arow * DIM;

    // ---- GEMM 1: (hvn + aggr) @ W0 ----
    v8f acc[6];
#pragma unroll
    for (int nt = 0; nt < 6; ++nt) acc[nt] = {};
    for (int kk = 0; kk < 24; ++kk) {
      int k = 4 * kk + koff;
      v2f a;
      a.x = ha[k] + ga[k];
      a.y = ha[k + 1] + ga[k + 1];
#pragma unroll
      for (int nt = 0; nt < 6; ++nt) {
        v2f b;
        b.x = W0lds[k * DIM + nt * 16 + m];
        b.y = W0lds[(k + 1) * DIM + nt * 16 + m];
        acc[nt] = wmma_f32(a, b, acc[nt]);
      }
    }

    // bias + relu, stage z (row-major, wave-private) for A-layout reload
#pragma unroll
    for (int nt = 0; nt < 6; ++nt) {
#pragma unroll
      for (int r = 0; r < 8; ++r) {
        float v = acc[nt][r] + b0c[nt];
        zs[wid][(r + half) * DIM + nt * 16 + m] = v > 0.0f ? v : 0.0f;
      }
    }

    // ---- GEMM 2: z @ W1 ----
#pragma unroll
    for (int nt = 0; nt < 6; ++nt) acc[nt] = {};
    for (int kk = 0; kk < 24; ++kk) {
      int k = 4 * kk + koff;
      v2f a;
      a.x = zs[wid][m * DIM + k];
      a.y = zs[wid][m * DIM + k + 1];
#pragma unroll
      for (int nt = 0; nt < 6; ++nt) {
        v2f b;
        b.x = W1lds[k * DIM + nt * 16 + m];
        b.y = W1lds[(k + 1) * DIM + nt * 16 + m];
        acc[nt] = wmma_f32(a, b, acc[nt]);
      }
    }

    // epilogue: write h_out; pooled sum into g (JK-cat slice = layer*96)
    if (n0 + 16 <= N) {  // wave-uniform fast path
#pragma unroll
      for (int r = 0; r < 8; ++r) {
        const int n  = n0 + r + half;
        const int bg = batch[n];
        float* hrow = hout + (size_t)n * DIM + m;
        float* grow = g + (size_t)bg * CATD + layer * DIM + m;
#pragma unroll
        for (int nt = 0; nt < 6; ++nt) {
          float v = acc[nt][r] + b1c[nt];
          hrow[nt * 16] = v;
          atomicAdd(&grow[nt * 16], v);
        }
      }
    } else {  // tail chunk
#pragma unroll
      for (int r = 0; r < 8; ++r) {
        const int n = n0 + r + half;
        if (n < N) {
          const int bg = batch[n];
          float* hrow = hout + (size_t)n * DIM + m;
          float* grow = g + (size_t)bg * CATD + layer * DIM + m;
#pragma unroll
          for (int nt = 0; nt < 6; ++nt) {
            float v = acc[nt][r] + b1c[nt];
            hrow[nt * 16] = v;
            atomicAdd(&grow[nt * 16], v);
          }
        }
      }
    }
  }
}

// ---------------------------------------------------------------------------
// Head MLP: out = relu(LN(row @ W + b)). One block per graph, Dout threads.
// ---------------------------------------------------------------------------
__global__ void mlp_ln_relu_kernel(const float* __restrict__ in,
                                   const float* __restrict__ W,
                                   const float* __restrict__ b,
                                   const float* __restrict__ gamma,
                                   const float* __restrict__ beta,
                                   float* __restrict__ out,
                                   int Din, int Dout) {
  __shared__ float red[256];
  const int row = blockIdx.x;
  const int t   = threadIdx.x;
  const float* rin = in + row * Din;

  float acc = b[t];
  for (int k = 0; k < Din; ++k) acc += rin[k] * W[k * Dout + t];

  red[t] = acc;
  __syncthreads();
  for (int s = Dout >> 1; s > 0; s >>= 1) {
    if (t < s) red[t] += red[t + s];
    __syncthreads();
  }
  float mu = red[0] / (float)Dout;
  __syncthreads();
  float d = acc - mu;
  red[t] = d * d;
  __syncthreads();
  for (int s = Dout >> 1; s > 0; s >>= 1) {
    if (t < s) red[t] += red[t + s];
    __syncthreads();
  }
  float var = red[0] / (float)Dout;
  float y = d * rsqrtf(var + 1e-5f) * gamma[t] + beta[t];
  out[row * Dout + t] = y > 0.0f ? y : 0.0f;
}

// emb = g2 @ W_emb + b_emb ; out = emb @ W_out + b_out   (one block per graph)
__global__ void head_kernel(const float* __restrict__ in,
                            const float* __restrict__ We,
                            const float* __restrict__ be,
                            const float* __restrict__ Wo,
                            const float* __restrict__ bo,
                            float* __restrict__ out) {
  __shared__ float emb[128];
  const int row = blockIdx.x;
  const int t   = threadIdx.x;
  const float* rin = in + row * 128;

  float acc = be[t];
  for (int k = 0; k < 128; ++k) acc += rin[k] * We[k * 128 + t];
  emb[t] = acc;
  __syncthreads();
  if (t < 12) {
    float o = bo[t];
    for (int k = 0; k < 128; ++k) o += emb[k] * Wo[k * 12 + t];
    out[row * 12 + t] = o;
  }
}

// ---------------------------------------------------------------------------
// Host-side orchestration
// ---------------------------------------------------------------------------
extern "C" void kernel_launch(void* const* d_in, const int* in_sizes, int n_in,
                              void* d_out, int out_size, void* d_ws,
                              size_t ws_size, hipStream_t stream) {
  // inputs in setup_inputs() insertion order
  const float* x     = (const float*)d_in[0];
  const int*   ei    = (const int*)d_in[1];
  const float* ea    = (const float*)d_in[2];
  const int*   batch = (const int*)d_in[3];
  const int N  = in_sizes[0] / DIM;
  const int E  = in_sizes[1] / 2;
  const int*   src = ei;
  const int*   dst = ei + E;

  const int P = 4;  // params base; 7 per layer
  auto F = [&](int i) { return (const float*)d_in[i]; };
  // per-layer: vn_emb, We, be, W0, b0, W1, b1
  // then: Wl_0,bl_0,lng_0,lnb_0, Wl_1,bl_1,lng_1,lnb_1, W_emb,b_emb,W_out,b_out

  // workspace layout (floats)
  float* ws   = (float*)d_ws;
  float* hvn  = ws;                      // N*96
  float* aggr = hvn + (size_t)N * DIM;   // N*96
  float* h    = aggr + (size_t)N * DIM;  // N*96 (layer output, overwritten)
  float* g    = h + (size_t)N * DIM;     // 512*288 pooled JK-cat
  float* g1   = g + (size_t)NGR * CATD;  // 512*256
  float* g2   = g1 + (size_t)NGR * 256;  // 512*128

  const int threads = 256;
  const int totND   = N * DIM;

  zero_kernel<<<(NGR * CATD + threads - 1) / threads, threads, 0, stream>>>(
      g, NGR * CATD);

  const int edgeTiles  = (E + 15) / 16;
  int edgeBlocks = (edgeTiles + 7) / 8;           // 8 waves per 256-thr block
  if (edgeBlocks > 1024) edgeBlocks = 1024;       // persistent waves
  const int nodeChunks = (N + 16 * NODE_WAVES - 1) / (16 * NODE_WAVES);
  int nodeBlocks = nodeChunks;
  if (nodeBlocks > 512) nodeBlocks = 512;         // persistent blocks

  for (int L = 0; L < 3; ++L) {
    const float* vn = F(P + 7 * L + 0);
    const float* We = F(P + 7 * L + 1);
    const float* be = F(P + 7 * L + 2);
    const float* W0 = F(P + 7 * L + 3);
    const float* b0 = F(P + 7 * L + 4);
    const float* W1 = F(P + 7 * L + 5);
    const float* b1 = F(P + 7 * L + 6);

    prep_kernel<<<(totND + threads - 1) / threads, threads, 0, stream>>>(
        L == 0 ? x : h, vn, hvn, aggr, totND);

    edge_kernel<<<edgeBlocks, 256, 0, stream>>>(ea, src, dst, We, be, hvn,
                                                aggr, E, edgeTiles);

    node_kernel<<<nodeBlocks, 32 * NODE_WAVES, 0, stream>>>(
        hvn, aggr, W0, b0, W1, b1, h, g, batch, L, N, nodeChunks);
  }

  // head MLP
  const float* Wl0 = F(P + 21), *bl0 = F(P + 22), *lg0 = F(P + 23),
             *lb0 = F(P + 24);
  const float* Wl1 = F(P + 25), *bl1 = F(P + 26), *lg1 = F(P + 27),
             *lb1 = F(P + 28);
  const float* Wemb = F(P + 29), *bemb = F(P + 30);
  const float* Wout = F(P + 31), *bout = F(P + 32);

  mlp_ln_relu_kernel<<<NGR, 256, 0, stream>>>(g, Wl0, bl0, lg0, lb0, g1, CATD,
                                              256);
  mlp_ln_relu_kernel<<<NGR, 128, 0, stream>>>(g1, Wl1, bl1, lg1, lb1, g2, 256,
                                              128);
  head_kernel<<<NGR, 128, 0, stream>>>(g2, Wemb, bemb, Wout, bout,
                                       (float*)d_out);
}